// WavefieldLoss_69320772157809
// MI455X (gfx1250) — compile-verified
//
#include <hip/hip_runtime.h>
#include <cstdint>

// W2 wavefield loss, fused single-pass design for MI455X (gfx1250).
// 163.8 MB total input, read exactly once via TDM tensor_load_to_lds into
// 160 KB of LDS per workgroup; per-trace CDF scan + two-pointer quantile
// merge run entirely on-chip. HBM-bound floor ~7us @ 23.3 TB/s.

#define NT     2500          // time samples per trace
#define NSEG   2499          // trapezoid segments
#define NR     256           // receivers
#define NSHOT  32
#define G      8             // traces (receivers) per workgroup
#define NGRP   (NR / G)      // 32 receiver groups per shot
#define THREADS 512          // 16 waves: 8 receivers x {syn,obs}
#define NTRACE (NSHOT * NR)  // 8192
#define DTX    0.001f
#define BNORM  30.0f

typedef unsigned int v4u __attribute__((ext_vector_type(4)));
typedef unsigned int v8u __attribute__((ext_vector_type(8)));

// Low 32 bits of a generic pointer to __shared__ == LDS byte offset (aperture rule).
__device__ __forceinline__ uint32_t lds_off(const void* p) {
  return (uint32_t)(uintptr_t)p;
}

// Issue one TDM 2D tile load: tile (G recv x NT time), element 4B,
// row stride NR elements, compacted row-major into LDS -> layout [t][G].
__device__ __forceinline__ void tdm_load_2d(uint32_t ldsb, const float* gptr) {
  uint64_t ga = (uint64_t)(uintptr_t)gptr;
  v4u g0;
  g0[0] = 1u;                                   // count=1 valid descriptor
  g0[1] = ldsb;                                 // lds_addr (bytes)
  g0[2] = (uint32_t)ga;                         // global_addr[31:0]
  g0[3] = ((uint32_t)(ga >> 32) & 0x1FFFFFFu)   // global_addr[56:32]
          | 0x80000000u;                        // type=2 ("image")
  v8u g1;
  g1[0] = 0x00020000u;                          // wg_mask=0, data_size=2 (4B)
  g1[1] = ((uint32_t)NR & 0xFFFFu) << 16;       // tensor_dim0 lo (256)
  g1[2] = ((uint32_t)NT & 0xFFFFu) << 16;       // tensor_dim1 lo (2500)
  g1[3] = ((uint32_t)G) << 16;                  // tile_dim0 = 8
  g1[4] = (uint32_t)NT;                         // tile_dim1 = 2500
  g1[5] = (uint32_t)NR;                         // tensor_dim0_stride lo = 256
  g1[6] = 0u;
  g1[7] = 0u;
  asm volatile("tensor_load_to_lds %0, %1" :: "s"(g0), "s"(g1) : "memory");
}

// Raw (unnormalized) CDF in LDS, interleaved: c[k] (k>=1) at slot (k-1)*G + r,
// c[0]==0 implicit. Find clip(searchsorted_left(c, tv), 1, NT-1) by binary
// search restricted to [1, NT] (tv<=0 correctly yields 1).
__device__ __forceinline__ int lower_bound_ri(const float* __restrict__ c,
                                              int r, float tv) {
  int lo = 1, hi = NT;
  while (lo < hi) {
    int mid = (lo + hi) >> 1;
    float v = c[(mid - 1) * G + r];
    if (v < tv) lo = mid + 1; else hi = mid;
  }
  return lo > NT - 1 ? NT - 1 : lo;
}

// Interpolated quantile given bracket [ri-1, ri] with raw CDF values cl, cr.
// s = A_tot + 1e-9 (normalization denominator).
__device__ __forceinline__ float qinterp(int ri, float cl, float cr,
                                         float tv, float s) {
  float d = cr - cl;                  // raw dcdf; normalized dcdf = d/s
  float base = DTX * (float)(ri - 1); // x[li]
  if (d < 1e-8f * s)                  // reference: dcdf<1e-8 -> use 1.0
    return base + ((tv - cl) / s) * DTX;
  return base + DTX * (tv - cl) / d;
}

__global__ __launch_bounds__(THREADS)
void w2_kernel(const float* __restrict__ syn, const float* __restrict__ obs,
               float* __restrict__ perTrace) {
  __shared__ float cbuf[2 * NT * G];   // 160,000 B: {syn,obs} x [t][G]
  __shared__ float wsum[16];

  const int shot = (int)blockIdx.x / NGRP;
  const int grp  = (int)blockIdx.x % NGRP;
  const int tid  = (int)threadIdx.x;
  const int lane = tid & 31;
  const int wave = tid >> 5;

  // ---- TDM stage: DMA both (G x NT) tiles into LDS (wave 0 issues) ----
  if (tid < 32) {
    const uint64_t base = (uint64_t)shot * NT * NR + (uint64_t)grp * G;
    tdm_load_2d(lds_off(&cbuf[0]),      syn + base);
    tdm_load_2d(lds_off(&cbuf[NT * G]), obs + base);
#if __has_builtin(__builtin_amdgcn_s_wait_tensorcnt)
    __builtin_amdgcn_s_wait_tensorcnt(0);
#else
    asm volatile("s_wait_tensorcnt 0x0" ::: "memory");
#endif
  }
  __syncthreads();

  // ---- Phase A: in-place raw-CDF scan. Wave w owns stream (r = w>>1,
  //      arr = w&1). seg[t] = ((p[t]+p[t+1])/2 + 30)*dt; slot t <- c[t+1].
  //      p[t+1] comes from the lane+1 neighbor via shuffle (boundary lane
  //      does the one extra LDS read). ----
  {
    const int r = wave >> 1;
    float* c = &cbuf[(wave & 1) * NT * G];
    float carry = 0.0f;
    for (int i0 = 0; i0 < NT; i0 += 32) {
      const int t = i0 + lane;
      const bool v = (t < NSEG);
      float pt = (t < NT) ? c[t * G + r] : 0.0f;
      float pn = __shfl_down(pt, 1, 32);               // p[t+1] for lanes 0..30
      if (lane == 31) pn = (t + 1 < NT) ? c[(t + 1) * G + r] : 0.0f;
      float s = v ? ((pt + pn) * 0.5f + BNORM) * DTX : 0.0f;
      // wave32 inclusive scan
      for (int off = 1; off < 32; off <<= 1) {
        float n = __shfl_up(s, off, 32);
        if (lane >= off) s += n;
      }
      float cv = carry + s;
      if (v) c[t * G + r] = cv;                        // p[t] dead -> store c[t+1]
      carry = __shfl(cv, 31, 32);
    }
  }
  __syncthreads();

  // ---- Phase B: two-pointer quantile merge + trapezoid integration.
  //      64 threads per trace; each owns a contiguous block of 40 quantile
  //      levels: one binary search to seed, then monotone marching. ----
  const int tr  = tid >> 6;            // 0..7
  const int sub = tid & 63;
  const float* c1 = &cbuf[0];
  const float* c2 = &cbuf[NT * G];
  const float s1 = c1[(NSEG - 1) * G + tr] + 1e-9f;  // A_tot = c[2499]
  const float s2 = c2[(NSEG - 1) * G + tr] + 1e-9f;

  const int per = (NT + 63) >> 6;      // 40 levels per thread
  const int k0  = sub * per;
  const int k1  = (k0 + per < NT) ? (k0 + per) : NT;

  float acc = 0.0f;
  if (k0 < NT) {
    const float tk0 = (float)k0 * (1.0f / (float)NSEG);
    int ri1 = lower_bound_ri(c1, tr, tk0 * s1);
    int ri2 = lower_bound_ri(c2, tr, tk0 * s2);
    float cr1 = c1[(ri1 - 1) * G + tr];
    float cl1 = (ri1 >= 2) ? c1[(ri1 - 2) * G + tr] : 0.0f;
    float cr2 = c2[(ri2 - 1) * G + tr];
    float cl2 = (ri2 >= 2) ? c2[(ri2 - 2) * G + tr] : 0.0f;

    for (int k = k0; k < k1; ++k) {
      float tk = (float)k * (1.0f / (float)NSEG);
      float tv1 = tk * s1;
      while (ri1 < NT - 1 && cr1 < tv1) {
        ++ri1; cl1 = cr1; cr1 = c1[(ri1 - 1) * G + tr];
      }
      float tv2 = tk * s2;
      while (ri2 < NT - 1 && cr2 < tv2) {
        ++ri2; cl2 = cr2; cr2 = c2[(ri2 - 1) * G + tr];
      }
      float q1 = qinterp(ri1, cl1, cr1, tv1, s1);
      float q2 = qinterp(ri2, cl2, cr2, tv2, s2);
      float d = q1 - q2;
      float w = d * d;
      acc += (k == 0 || k == NT - 1) ? 0.5f * w : w;   // trapezoid endpoints
    }
  }
  // reduce within wave, then across the 2 waves of this trace
  for (int off = 16; off; off >>= 1) acc += __shfl_down(acc, off, 32);
  if (lane == 0) wsum[wave] = acc;
  __syncthreads();
  if (tid < G) {
    float tot = wsum[2 * tid] + wsum[2 * tid + 1];
    float w2  = sqrtf(tot * (1.0f / (float)NSEG));
    perTrace[((uint32_t)blockIdx.x) * G + tid] = w2;
  }
}

__global__ __launch_bounds__(256)
void mean_kernel(const float* __restrict__ perTrace, float* __restrict__ out) {
  __shared__ float sm[8];
  float a = 0.0f;
  for (int i = (int)threadIdx.x; i < NTRACE; i += 256) a += perTrace[i];
  for (int off = 16; off; off >>= 1) a += __shfl_down(a, off, 32);
  if ((threadIdx.x & 31) == 0) sm[threadIdx.x >> 5] = a;
  __syncthreads();
  if (threadIdx.x == 0) {
    float t = 0.0f;
    for (int i = 0; i < 8; ++i) t += sm[i];
    out[0] = t * (1.0f / (float)NTRACE);
  }
}

extern "C" void kernel_launch(void* const* d_in, const int* in_sizes, int n_in,
                              void* d_out, int out_size, void* d_ws, size_t ws_size,
                              hipStream_t stream) {
  const float* syn = (const float*)d_in[0];
  const float* obs = (const float*)d_in[1];
  float* per = (float*)d_ws;            // needs 8192 floats (32 KB) of scratch
  float* out = (float*)d_out;
  hipLaunchKernelGGL(w2_kernel, dim3(NSHOT * NGRP), dim3(THREADS), 0, stream,
                     syn, obs, per);
  hipLaunchKernelGGL(mean_kernel, dim3(1), dim3(256), 0, stream, per, out);
}